// ContrastiveLoss_6339371729129
// MI455X (gfx1250) — compile-verified
//
#include <hip/hip_runtime.h>

typedef __attribute__((ext_vector_type(2))) float v2f;
typedef __attribute__((ext_vector_type(4))) float v4f;
typedef __attribute__((ext_vector_type(8))) float v8f;

#define ALPHA_C 0.1f
#define BETA_C 0.3f
#define MARGIN_C 2.0f
#define LROWS 2048
#define DDIM 512
#define NBATCH 8

// ---------------------------------------------------------------------------
// Kernel 1: row squared-norms. One wave (32 lanes) per row of 512 floats.
// ---------------------------------------------------------------------------
__global__ __launch_bounds__(256) void norms_kernel(const float* __restrict__ x,
                                                    float* __restrict__ norms) {
    const int wave = threadIdx.x >> 5;
    const int lane = threadIdx.x & 31;
    const int row  = blockIdx.x * 8 + wave;          // 0 .. 16383
    const float* p = x + (size_t)row * DDIM;
    float s = 0.f;
#pragma unroll
    for (int c = 0; c < 4; ++c) {
        v4f v = *(const v4f*)(p + c * 128 + lane * 4);
        s += v.x * v.x + v.y * v.y + v.z * v.z + v.w * v.w;
    }
#pragma unroll
    for (int m = 16; m >= 1; m >>= 1) s += __shfl_xor(s, m, 32);
    if (lane == 0) norms[row] = s;
}

// ---------------------------------------------------------------------------
// Epilogue for one 16x16 WMMA accumulator tile.
// C/D layout: element r of the v8f <-> M = rowbase + r (rowbase already
// includes the +8*half for lanes 16..31), N = lane & 15 (sqC/segC).
// ---------------------------------------------------------------------------
__device__ inline float tile_loss(v8f acc, const float* __restrict__ nrm,
                                  const int* __restrict__ sg, int rowbase,
                                  float sqC, int segC) {
    float s = 0.f;
#pragma unroll
    for (int r = 0; r < 8; ++r) {
        const int row = rowbase + r;
        float d2 = nrm[row] + sqC - 2.0f * acc[r];
        d2 = fmaxf(d2, 0.f);
        const float dist = __builtin_sqrtf(d2);
        const float h = fmaxf(MARGIN_C - dist, 0.f);
        s += (sg[row] == segC) ? (ALPHA_C * d2) : (BETA_C * h * h);
    }
    return s;
}

// ---------------------------------------------------------------------------
// Kernel 2: fused Gram-tile GEMM (v_wmma_f32_16x16x4_f32) + loss epilogue +
// block reduction. Block = 256 threads = 8 waves; each wave computes a
// 32x32 output region (2x2 WMMA tiles); block covers 128(j) x 64(k).
// Grid = 8 batches * 16 * 32 = 4096 blocks.
//
// Load scheme: per lane, one b128 load at (row*512 + k0 + 4*half) feeds TWO
// WMMAs (.xy then .zw). Hardware K-slot = 2*half + v for both A and B f32
// fragments; using the same memory<->slot permutation for A and B keeps the
// K-sum exact while covering k0..k0+7 once per float4. This gives a 1:2
// vmem:wmma instruction ratio instead of 1:1.
// ---------------------------------------------------------------------------
__global__ __launch_bounds__(256) void tile_kernel(const float* __restrict__ x,
                                                   const int* __restrict__ seg,
                                                   const float* __restrict__ norms,
                                                   float* __restrict__ partials) {
    const int wave = threadIdx.x >> 5;
    const int lane = threadIdx.x & 31;
    const int n    = lane & 15;
    const int half = lane >> 4;

    const int bid   = blockIdx.x;
    const int batch = bid >> 9;          // / 512 blocks per batch
    const int rem   = bid & 511;
    const int bj    = rem >> 5;          // 0..15  (j block of 128)
    const int bk    = rem & 31;          // 0..31  (k block of 64)
    const int jbase = bj * 128 + (wave & 3) * 32;
    const int kbase = bk * 64 + (wave >> 2) * 32;

    const float* X = x + (size_t)batch * LROWS * DDIM;
    // Each lane's float4 covers memory ks [k0+4*half, k0+4*half+3].
    const float* rA0 = X + (size_t)(jbase + n) * DDIM + 4 * half;
    const float* rA1 = rA0 + 16 * DDIM;
    const float* rB0 = X + (size_t)(kbase + n) * DDIM + 4 * half;
    const float* rB1 = rB0 + 16 * DDIM;

    v8f acc00 = {}; v8f acc01 = {}; v8f acc10 = {}; v8f acc11 = {};

#pragma unroll 2
    for (int k0 = 0; k0 < DDIM; k0 += 8) {
        v4f a0 = *(const v4f*)(rA0 + k0);
        v4f a1 = *(const v4f*)(rA1 + k0);
        v4f b0 = *(const v4f*)(rB0 + k0);
        v4f b1 = *(const v4f*)(rB1 + k0);

        v2f a0l = a0.xy, a0h = a0.zw;
        v2f a1l = a1.xy, a1h = a1.zw;
        v2f b0l = b0.xy, b0h = b0.zw;
        v2f b1l = b1.xy, b1h = b1.zw;

        acc00 = __builtin_amdgcn_wmma_f32_16x16x4_f32(false, a0l, false, b0l,
                                                      (short)0, acc00, false, false);
        acc01 = __builtin_amdgcn_wmma_f32_16x16x4_f32(false, a0l, false, b1l,
                                                      (short)0, acc01, false, false);
        acc10 = __builtin_amdgcn_wmma_f32_16x16x4_f32(false, a1l, false, b0l,
                                                      (short)0, acc10, false, false);
        acc11 = __builtin_amdgcn_wmma_f32_16x16x4_f32(false, a1l, false, b1l,
                                                      (short)0, acc11, false, false);

        acc00 = __builtin_amdgcn_wmma_f32_16x16x4_f32(false, a0h, false, b0h,
                                                      (short)0, acc00, false, false);
        acc01 = __builtin_amdgcn_wmma_f32_16x16x4_f32(false, a0h, false, b1h,
                                                      (short)0, acc01, false, false);
        acc10 = __builtin_amdgcn_wmma_f32_16x16x4_f32(false, a1h, false, b0h,
                                                      (short)0, acc10, false, false);
        acc11 = __builtin_amdgcn_wmma_f32_16x16x4_f32(false, a1h, false, b1h,
                                                      (short)0, acc11, false, false);
    }

    const float* nrm = norms + batch * LROWS;
    const int* sg = seg + batch * LROWS;
    const float sqC0 = nrm[kbase + n];
    const float sqC1 = nrm[kbase + 16 + n];
    const int segC0 = sg[kbase + n];
    const int segC1 = sg[kbase + 16 + n];

    float lsum = 0.f;
    lsum += tile_loss(acc00, nrm, sg, jbase + 8 * half,      sqC0, segC0);
    lsum += tile_loss(acc01, nrm, sg, jbase + 8 * half,      sqC1, segC1);
    lsum += tile_loss(acc10, nrm, sg, jbase + 16 + 8 * half, sqC0, segC0);
    lsum += tile_loss(acc11, nrm, sg, jbase + 16 + 8 * half, sqC1, segC1);

#pragma unroll
    for (int m = 16; m >= 1; m >>= 1) lsum += __shfl_xor(lsum, m, 32);

    __shared__ float wsum[8];
    if (lane == 0) wsum[wave] = lsum;
    __syncthreads();
    if (threadIdx.x == 0) {
        float t = 0.f;
#pragma unroll
        for (int i = 0; i < 8; ++i) t += wsum[i];
        partials[bid] = t;
    }
}

// ---------------------------------------------------------------------------
// Kernel 3: deterministic final reduction of 4096 partials -> scalar mean.
// ---------------------------------------------------------------------------
__global__ __launch_bounds__(256) void reduce_kernel(const float* __restrict__ partials,
                                                     int nPart, float* __restrict__ out) {
    __shared__ float sm[256];
    float s = 0.f;
    for (int i = threadIdx.x; i < nPart; i += 256) s += partials[i];
    sm[threadIdx.x] = s;
    __syncthreads();
    for (int st = 128; st > 0; st >>= 1) {
        if (threadIdx.x < st) sm[threadIdx.x] += sm[threadIdx.x + st];
        __syncthreads();
    }
    if (threadIdx.x == 0)
        out[0] = sm[0] * (1.0f / ((float)NBATCH * (float)LROWS * (float)LROWS));
}

extern "C" void kernel_launch(void* const* d_in, const int* in_sizes, int n_in,
                              void* d_out, int out_size, void* d_ws, size_t ws_size,
                              hipStream_t stream) {
    const float* pred = (const float*)d_in[0];           // [8, 2048, 512] fp32
    const int*   seg  = (const int*)d_in[1];             // [8, 2048] int
    float* out = (float*)d_out;                          // scalar

    float* norms    = (float*)d_ws;                      // 16384 floats
    float* partials = norms + NBATCH * LROWS;            // 4096 floats

    // 16384 rows / 8 waves-per-block = 2048 blocks
    norms_kernel<<<2048, 256, 0, stream>>>(pred, norms);
    // 8 batches * (2048/128) * (2048/64) = 4096 blocks
    tile_kernel<<<4096, 256, 0, stream>>>(pred, seg, norms, partials);
    reduce_kernel<<<1, 256, 0, stream>>>(partials, 4096, out);
}